// ElectrostaticEnergyLayer_59992103190615
// MI455X (gfx1250) — compile-verified
//
#include <hip/hip_runtime.h>

// ---------------------------------------------------------------------------
// Electrostatic pair-energy + segment-sum (PhysNet shielded Coulomb).
//   E(r) = KEHALF*Qi*Qj*( sw(r)*E_sh(r) + (1-sw(r))*E_ord(r) ),  r <= 10 else 0
//   scatter-add E into out[idx_i].
//
// MI455X reasoning: 12 B/pair streamed (Dij + idx_i + idx_j) -> ~201 MB at
// 23.3 TB/s = ~9 us floor. Qa (1 MB) and out (1 MB) are hit ~47M times and
// must stay L2-resident, so the read-once streams are loaded with th:NT
// (CDNA5 temporal hint) to avoid cycling the 192 MB L2. Scatter-add uses
// native global_atomic_add_f32 (no-return, STOREcnt-tracked, L2 atomic units),
// predicated on r <= LR_CUTOFF which removes ~17.5% of atomics exactly.
// ---------------------------------------------------------------------------

typedef float f32x4 __attribute__((ext_vector_type(4)));
typedef int   i32x4 __attribute__((ext_vector_type(4)));

namespace {

constexpr float KEHALF      = 7.1998226759752740f;
constexpr float LR_CUTOFF   = 10.0f;
constexpr float INV_LR2     = 1.0f / (LR_CUTOFF * LR_CUTOFF);   // 0.01
constexpr float TWO_OVER_LR = 2.0f / LR_CUTOFF;                 // 0.2
constexpr float INV_SR_HALF = 0.5f;   // 1 / (SR_CUTOFF/2) with SR_CUTOFF = 4

constexpr int PAIRS_PER_THREAD = 4;
constexpr int BLOCK            = 256;   // 8 wave32 per workgroup

__device__ __forceinline__ float pair_energy(float r, float qq) {
    // PhysNet smooth switch: phi(2r / SR_CUTOFF), x = r/2
    float x  = r * INV_SR_HALF;
    float sw = 0.0f;
    if (x < 1.0f) {
        float x2 = x * x;
        // 1 - x^3 * (10 - 15x + 6x^2)
        sw = 1.0f - (x2 * x) * fmaf(x, fmaf(6.0f, x, -15.0f), 10.0f);
    }
    float rsh   = sqrtf(fmaf(r, r, 1.0f));                       // shielded dist
    float e_ord = 1.0f / r   + fmaf(r,   INV_LR2, -TWO_OVER_LR);
    float e_sh  = 1.0f / rsh + fmaf(rsh, INV_LR2, -TWO_OVER_LR);
    float mix   = sw * e_sh + (1.0f - sw) * e_ord;
    return KEHALF * qq * mix;
}

// Defined first so the disasm snippet shows this kernel's body
// (verify: global_load_b128 ... th:TH_LOAD_NT and global_atomic_add_f32).
__global__ __launch_bounds__(BLOCK) void ele_energy_kernel(
    const float* __restrict__ Dij,
    const float* __restrict__ Qa,
    const int*   __restrict__ idx_i,
    const int*   __restrict__ idx_j,
    float*       __restrict__ out,
    int n_pairs) {
    const int tid  = blockIdx.x * blockDim.x + threadIdx.x;
    const int base = tid * PAIRS_PER_THREAD;

    if (base + PAIRS_PER_THREAD <= n_pairs) {
        // Read-once streams: non-temporal b128 loads keep Qa/out L2-resident.
        const f32x4 d =
            __builtin_nontemporal_load(reinterpret_cast<const f32x4*>(Dij + base));
        const i32x4 ii =
            __builtin_nontemporal_load(reinterpret_cast<const i32x4*>(idx_i + base));
        const i32x4 jj =
            __builtin_nontemporal_load(reinterpret_cast<const i32x4*>(idx_j + base));

        // Gather charges (hot 1 MB table, default RT policy -> L2 hits).
        float qq[PAIRS_PER_THREAD];
#pragma unroll
        for (int k = 0; k < PAIRS_PER_THREAD; ++k) qq[k] = Qa[ii[k]] * Qa[jj[k]];

#pragma unroll
        for (int k = 0; k < PAIRS_PER_THREAD; ++k) {
            // r > LR_CUTOFF contributes exactly 0 -> skip the atomic entirely.
            if (d[k] <= LR_CUTOFF) {
                atomicAdd(out + ii[k], pair_energy(d[k], qq[k]));
            }
        }
    } else {
        // Scalar tail (not exercised for N_PAIRS = 2^24 but kept generic).
        for (int p = base; p < n_pairs; ++p) {
            float r = Dij[p];
            if (r <= LR_CUTOFF) {
                float qq = Qa[idx_i[p]] * Qa[idx_j[p]];
                atomicAdd(out + idx_i[p], pair_energy(r, qq));
            }
        }
    }
}

__global__ __launch_bounds__(BLOCK) void zero_out_kernel(float* __restrict__ p,
                                                         int n) {
    int t = (blockIdx.x * blockDim.x + threadIdx.x) * 4;
    if (t + 4 <= n) {
        *reinterpret_cast<f32x4*>(p + t) = f32x4{0.0f, 0.0f, 0.0f, 0.0f};
    } else {
        for (int k = t; k < n; ++k) p[k] = 0.0f;
    }
}

}  // namespace

extern "C" void kernel_launch(void* const* d_in, const int* in_sizes, int n_in,
                              void* d_out, int out_size, void* d_ws,
                              size_t ws_size, hipStream_t stream) {
    const float* Dij   = reinterpret_cast<const float*>(d_in[0]);
    const float* Qa    = reinterpret_cast<const float*>(d_in[1]);
    const int*   idx_i = reinterpret_cast<const int*>(d_in[2]);
    const int*   idx_j = reinterpret_cast<const int*>(d_in[3]);
    float*       out   = reinterpret_cast<float*>(d_out);

    const int n_pairs = in_sizes[0];
    const int n_atoms = out_size;

    // 1) Zero the (poisoned) output accumulator.
    {
        int threads = (n_atoms + 3) / 4;
        int blocks  = (threads + BLOCK - 1) / BLOCK;
        zero_out_kernel<<<blocks, BLOCK, 0, stream>>>(out, n_atoms);
    }

    // 2) Pair energies + scatter-add.
    {
        int threads = (n_pairs + PAIRS_PER_THREAD - 1) / PAIRS_PER_THREAD;
        int blocks  = (threads + BLOCK - 1) / BLOCK;
        ele_energy_kernel<<<blocks, BLOCK, 0, stream>>>(Dij, Qa, idx_i, idx_j,
                                                        out, n_pairs);
    }
}